// Attention_20495583937178
// MI455X (gfx1250) — compile-verified
//
#include <hip/hip_runtime.h>

#define S_DIM 4096
#define B_DIM 8
#define T_DIM 512
#define H_DIM 400

typedef float v2f __attribute__((ext_vector_type(2)));
typedef float v8f __attribute__((ext_vector_type(8)));

#if defined(__AMDGCN__) && __has_builtin(__builtin_amdgcn_global_load_async_to_lds_b128)
#define HAS_ASYNC_LDS 1
#else
#define HAS_ASYNC_LDS 0
#endif

// 128-bit payload type the async builtin expects (GCC-style vector of 4 ints),
// in the global (1) and LDS (3) address spaces.
typedef int v4i __attribute__((vector_size(16)));
typedef __attribute__((address_space(1))) v4i as1_v4i;
typedef __attribute__((address_space(3))) v4i as3_v4i;

// Copy 16 contiguous bytes global -> LDS. Async (ASYNCcnt-tracked) when the
// gfx1250 async-to-LDS path is available; plain float4 round-trip otherwise.
__device__ __forceinline__ void cp16(const float* __restrict__ g, float* l) {
#if HAS_ASYNC_LDS
    // generic LDS pointers carry the LDS byte offset in their low 32 bits,
    // so truncating to a 32-bit addrspace(3) pointer is exact.
    __builtin_amdgcn_global_load_async_to_lds_b128(
        (as1_v4i*)(unsigned long long)g,
        (as3_v4i*)(unsigned)(unsigned long long)l,
        0, 0);
#else
    *(float4*)l = *(const float4*)g;
#endif
}

__device__ __forceinline__ void wait_async_lds() {
#if HAS_ASYNC_LDS
#if __has_builtin(__builtin_amdgcn_s_wait_asynccnt)
    __builtin_amdgcn_s_wait_asynccnt(0);
#else
    asm volatile("s_wait_asynccnt 0x0" ::: "memory");
#endif
#endif
}

// ---------------------------------------------------------------------------
// Kernel 1: scores[b,t,s] = dec[t,b,:] . enc[s,b,:], masked with -inf.
// grid (S/128, T/64, B), block 256 = 8 waves. Wave grid 2(m) x 4(n),
// each wave computes a 32x32 patch = 2x2 WMMA f32 16x16x4 accumulators.
// Double-buffered async LDS staging over K (H=400, 25 chunks of 16).
// ---------------------------------------------------------------------------
__global__ __launch_bounds__(256)
void scores_kernel(const float* __restrict__ enc,   // [S,B,H]
                   const float* __restrict__ dec,   // [T,B,H]
                   const int*   __restrict__ lens,  // [B]
                   float* __restrict__ scores)      // [B,T,S]
{
    __shared__ float At[2][64][20];    // [buf][t][k], stride 20 (16B-aligned rows)
    __shared__ float Bt[2][128][20];   // [buf][s][k]

    const int b    = blockIdx.z;
    const int t0   = blockIdx.y * 64;
    const int s0   = blockIdx.x * 128;
    const int tid  = threadIdx.x;
    const int lane = tid & 31;
    const int wave = tid >> 5;
    const int wm   = wave >> 2;      // 0..1
    const int wn   = wave & 3;       // 0..3
    const int half = lane >> 4;      // lanes 16..31 carry K={2,3}
    const int ln   = lane & 15;

    const int arow = tid >> 2;          // 0..63
    const int ac4  = (tid & 3) << 2;    // 0,4,8,12
    const float* agp = dec + ((size_t)(t0 + arow) * B_DIM + b) * H_DIM + ac4;
    const float* bgp = enc + ((size_t)(s0 + arow) * B_DIM + b) * H_DIM + ac4;
    const size_t brow_step = (size_t)64 * B_DIM * H_DIM;

    v8f acc[2][2];
    #pragma unroll
    for (int i = 0; i < 2; ++i)
        #pragma unroll
        for (int j = 0; j < 2; ++j)
            acc[i][j] = (v8f){};

    // prefetch chunk 0 into buffer 0
    cp16(agp, &At[0][arow][ac4]);
    cp16(bgp, &Bt[0][arow][ac4]);
    cp16(bgp + brow_step, &Bt[0][arow + 64][ac4]);

    int buf = 0;
    for (int k0 = 0; k0 < H_DIM; k0 += 16) {   // 25 exact chunks
        wait_async_lds();
        __syncthreads();
        if (k0 + 16 < H_DIM) {                 // overlap next chunk's staging
            const int kn = k0 + 16;
            cp16(agp + kn, &At[buf ^ 1][arow][ac4]);
            cp16(bgp + kn, &Bt[buf ^ 1][arow][ac4]);
            cp16(bgp + kn + brow_step, &Bt[buf ^ 1][arow + 64][ac4]);
        }

        #pragma unroll
        for (int kk = 0; kk < 16; kk += 4) {
            v2f a[2], bb[2];
            #pragma unroll
            for (int i = 0; i < 2; ++i) {
                const int rm = wm * 32 + i * 16 + ln;      // A: lane = M row
                a[i].x = At[buf][rm][kk + 2 * half];
                a[i].y = At[buf][rm][kk + 2 * half + 1];
            }
            #pragma unroll
            for (int j = 0; j < 2; ++j) {
                const int rn = wn * 32 + j * 16 + ln;      // B: lane = N col
                bb[j].x = Bt[buf][rn][kk + 2 * half];
                bb[j].y = Bt[buf][rn][kk + 2 * half + 1];
            }
            #pragma unroll
            for (int i = 0; i < 2; ++i)
                #pragma unroll
                for (int j = 0; j < 2; ++j)
                    acc[i][j] = __builtin_amdgcn_wmma_f32_16x16x4_f32(
                        false, a[i], false, bb[j], (short)0, acc[i][j],
                        false, false);
        }
        buf ^= 1;
    }

    const int len_b = lens[b];
    #pragma unroll
    for (int i = 0; i < 2; ++i) {
        #pragma unroll
        for (int j = 0; j < 2; ++j) {
            const int sg = s0 + wn * 32 + j * 16 + ln;
            const bool valid = sg < len_b;
            #pragma unroll
            for (int r = 0; r < 8; ++r) {
                const int tg = t0 + wm * 32 + i * 16 + r + 8 * half;
                const float v = valid ? acc[i][j][r] : -__builtin_inff();
                scores[((size_t)b * T_DIM + tg) * S_DIM + sg] = v;
            }
        }
    }
}

// ---------------------------------------------------------------------------
// Kernel 2: row softmax over S=4096, in place. One 256-thread WG per row.
// ---------------------------------------------------------------------------
__global__ __launch_bounds__(256)
void softmax_kernel(float* __restrict__ w)   // [B*T, S]
{
    __shared__ float red[256];
    const int tid = threadIdx.x;
    float* p = w + (size_t)blockIdx.x * S_DIM;

    float vals[16];
    float m = -__builtin_inff();
    #pragma unroll
    for (int i = 0; i < 16; ++i) {
        vals[i] = p[tid + i * 256];
        m = fmaxf(m, vals[i]);
    }
    red[tid] = m;
    __syncthreads();
    for (int off = 128; off > 0; off >>= 1) {
        if (tid < off) red[tid] = fmaxf(red[tid], red[tid + off]);
        __syncthreads();
    }
    m = red[0];
    __syncthreads();

    float s = 0.0f;
    #pragma unroll
    for (int i = 0; i < 16; ++i) {
        vals[i] = __expf(vals[i] - m);   // exp(-inf) = 0 handles mask
        s += vals[i];
    }
    red[tid] = s;
    __syncthreads();
    for (int off = 128; off > 0; off >>= 1) {
        if (tid < off) red[tid] += red[tid + off];
        __syncthreads();
    }
    const float inv = 1.0f / red[0];
    #pragma unroll
    for (int i = 0; i < 16; ++i)
        p[tid + i * 256] = vals[i] * inv;
}

// ---------------------------------------------------------------------------
// Kernel 3: context[b,t,h] = sum_s W[b,t,s] * enc[s,b,h].
// grid (T/64, B), block 256 = 8 waves. Wave w: m-tile rows (w>>1)*16,
// h-tiles nt = (w&1) + 2*j for j<13 (25 real tiles; phantom 26th lands in
// padded LDS columns and is never stored). Kc=16, double-buffered async LDS.
// ---------------------------------------------------------------------------
__global__ __launch_bounds__(256)
void context_kernel(const float* __restrict__ enc,  // [S,B,H]
                    const float* __restrict__ w,    // [B,T,S]
                    float* __restrict__ ctx)        // [B,T,H]
{
    __shared__ float At[2][64][20];     // [buf][t][k], Kc=16
    __shared__ float Bt[2][16][420];    // [buf][k][h], width 416 + 16B-align pad

    const int b    = blockIdx.y;
    const int t0   = blockIdx.x * 64;
    const int tid  = threadIdx.x;
    const int lane = tid & 31;
    const int wave = tid >> 5;
    const int mrow = (wave >> 1) * 16;
    const int ny   = wave & 1;
    const int half = lane >> 4;
    const int ln   = lane & 15;

    const int arow = tid >> 2;          // 0..63
    const int ac4  = (tid & 3) << 2;    // 0,4,8,12
    const float* agp = w + ((size_t)b * T_DIM + t0 + arow) * S_DIM + ac4;

    v8f acc[13];
    #pragma unroll
    for (int j = 0; j < 13; ++j) acc[j] = (v8f){};

    // prefetch chunk 0 into buffer 0
    cp16(agp, &At[0][arow][ac4]);
    for (int idx = tid; idx < 16 * 100; idx += 256) {
        const int row = idx / 100;
        const int c4  = (idx % 100) << 2;
        cp16(enc + ((size_t)row * B_DIM + b) * H_DIM + c4, &Bt[0][row][c4]);
    }

    int buf = 0;
    for (int k0 = 0; k0 < S_DIM; k0 += 16) {   // 256 chunks over S
        wait_async_lds();
        __syncthreads();
        if (k0 + 16 < S_DIM) {                 // overlap next chunk's staging
            const int kn = k0 + 16;
            cp16(agp + kn, &At[buf ^ 1][arow][ac4]);
            for (int idx = tid; idx < 16 * 100; idx += 256) {
                const int row = idx / 100;
                const int c4  = (idx % 100) << 2;
                cp16(enc + ((size_t)(kn + row) * B_DIM + b) * H_DIM + c4,
                     &Bt[buf ^ 1][row][c4]);
            }
        }

        #pragma unroll
        for (int kk = 0; kk < 16; kk += 4) {
            v2f a;
            a.x = At[buf][mrow + ln][kk + 2 * half];
            a.y = At[buf][mrow + ln][kk + 2 * half + 1];
            #pragma unroll
            for (int j = 0; j < 13; ++j) {
                const int hbase = (ny + 2 * j) * 16;
                v2f bb;
                bb.x = Bt[buf][kk + 2 * half][hbase + ln];
                bb.y = Bt[buf][kk + 2 * half + 1][hbase + ln];
                acc[j] = __builtin_amdgcn_wmma_f32_16x16x4_f32(
                    false, a, false, bb, (short)0, acc[j], false, false);
            }
        }
        buf ^= 1;
    }

    #pragma unroll
    for (int j = 0; j < 13; ++j) {
        const int hg = (ny + 2 * j) * 16 + ln;
        if (hg < H_DIM) {
            #pragma unroll
            for (int r = 0; r < 8; ++r) {
                const int tg = t0 + mrow + r + 8 * half;
                ctx[((size_t)b * T_DIM + tg) * H_DIM + hg] = acc[j][r];
            }
        }
    }
}

extern "C" void kernel_launch(void* const* d_in, const int* in_sizes, int n_in,
                              void* d_out, int out_size, void* d_ws, size_t ws_size,
                              hipStream_t stream) {
    const float* enc  = (const float*)d_in[0];   // [S,B,H]
    const float* dec  = (const float*)d_in[1];   // [T,B,H]
    const int*   lens = (const int*)d_in[2];     // [B]
    float* ctx = (float*)d_out;                                   // [B,T,H]
    float* wts = (float*)d_out + (size_t)B_DIM * T_DIM * H_DIM;   // [B,T,S]

    scores_kernel<<<dim3(S_DIM / 128, T_DIM / 64, B_DIM), 256, 0, stream>>>(
        enc, dec, lens, wts);
    softmax_kernel<<<dim3(B_DIM * T_DIM), 256, 0, stream>>>(wts);
    context_kernel<<<dim3(T_DIM / 64, B_DIM), 256, 0, stream>>>(
        enc, wts, ctx);
}